// ComplexGabor1DLayer_36481452212333
// MI455X (gfx1250) — compile-verified
//
#include <hip/hip_runtime.h>
#include <math.h>

// ---------------------------------------------------------------------------
// ComplexGabor1DLayer on MI455X (gfx1250, wave32, WMMA)
//   lin = x @ W^T + b            [N=262144, K=256, OUT=256]  (fp32 WMMA 16x16x4)
//   out[n,o] = exp(-3600*lin^2) * {cos(30*lin), sin(30*lin)} interleaved
//
// Structure (v2, anti-spill):
//   - Each workgroup: 128 rows x 128 output cols (colHalf = blockIdx.x & 1).
//   - Half of W^T staged in LDS (147456 B) -> 2 workgroups per WGP.
//   - Each wave: one 16-row tile x 8 col-tiles -> 64 accumulator VGPRs (no spill).
//   - Adjacent block pairs share the same x rows -> 2nd read hits L2.
// ---------------------------------------------------------------------------

typedef float v2f __attribute__((ext_vector_type(2)));
typedef float v8f __attribute__((ext_vector_type(8)));

#define K_DIM       256
#define OUT_DIM     256
#define OMEGA0      30.0f
#define SCALE0_SQ   3600.0f      // 60^2

#define WAVES_PER_WG  8
#define ROWS_PER_WAVE 16
#define ROWS_PER_WG   (WAVES_PER_WG * ROWS_PER_WAVE)   // 128
#define COLS_PER_WG   128
#define TILES_PER_WAVE (COLS_PER_WG / 16)              // 8

// LDS layout for the workgroup's half of W^T: element (k, oc) lives at
//   lds[ ((k>>1) * PAIR_STRIDE + oc) * 2 + (k & 1) ]      (oc = local col 0..127)
// PAIR_STRIDE = 144: 144*8 B = 1152 B = 128 mod 256 -> the two lane-halves of a
// wave (reading adjacent K-pair rows) hit disjoint halves of the 64 LDS banks.
#define PAIR_STRIDE 144
#define LDS_FLOATS  ((K_DIM / 2) * PAIR_STRIDE * 2)        // 36864 floats
#define LDS_BYTES   (LDS_FLOATS * sizeof(float))           // 147456 B -> 2 WGs/WGP

__global__ void __launch_bounds__(256, 2)
gabor_wmma_kernel(const float* __restrict__ x,
                  const float* __restrict__ W,
                  const float* __restrict__ bias,
                  float* __restrict__ out) {
    extern __shared__ float lds[];

    const int tid  = threadIdx.x;
    const int lane = tid & 31;
    const int wave = tid >> 5;
    const int hg   = lane >> 4;     // lane half-group (0: lanes 0-15, 1: 16-31)
    const int l15  = lane & 15;

    const int rowBlock = blockIdx.x >> 1;
    const int colBase  = (blockIdx.x & 1) * COLS_PER_WG;

    // ---- Stage this WG's half of W^T into LDS, K-pair interleaved --------
    // 256 threads cover 128 W rows; two threads per row split the K walk.
    // LDS writes are 8B-strided across lanes -> conflict-free.
    {
        const int oc    = tid & 127;              // local output col 0..127
        const int khalf = tid >> 7;               // 0 or 1
        const float4* wrow =
            (const float4*)(W + (size_t)(colBase + oc) * K_DIM) + khalf * (K_DIM / 8);
        #pragma unroll 4
        for (int i = 0; i < K_DIM / 8; ++i) {     // 32 float4s each
            float4 w = wrow[i];
            const int k = khalf * (K_DIM / 2) + i * 4;
            const int p = k >> 1;                 // k-pair index
            v2f lo = {w.x, w.y};
            v2f hi = {w.z, w.w};
            *(v2f*)&lds[((p    ) * PAIR_STRIDE + oc) * 2] = lo;
            *(v2f*)&lds[((p + 1) * PAIR_STRIDE + oc) * 2] = hi;
        }
    }
    __syncthreads();

    // ---- WMMA main loop ---------------------------------------------------
    const int rowBase = rowBlock * ROWS_PER_WG + wave * ROWS_PER_WAVE;

    // 8 col-tiles of accumulators (64 VGPRs), seeded with the bias.
    v8f acc[TILES_PER_WAVE];
    #pragma unroll
    for (int t = 0; t < TILES_PER_WAVE; ++t) {
        const float bv = bias[colBase + t * 16 + l15];
        v8f a0 = {bv, bv, bv, bv, bv, bv, bv, bv};
        acc[t] = a0;
    }

    // A fragment (16x4 fp32): lane L<16 holds row L, K = k0 + {0,1};
    // lane L>=16 holds row L-16, K = k0 + {2,3}.  -> per-lane float2 load.
    const float* aptr = x + ((size_t)(rowBase + l15)) * K_DIM + 2 * hg;

    #pragma unroll 2
    for (int k0 = 0; k0 < K_DIM; k0 += 4) {
        v2f afrag = *(const v2f*)(aptr + k0);

        // B fragment (4x16 fp32): VGPR j, lane L -> K = k0 + j + 2*(L>>4),
        // N = L&15.  Pair-interleaved LDS -> both VGPRs in one ds_load_b64.
        const int    kk   = k0 + 2 * hg;                 // even
        const float* brow = &lds[((kk >> 1) * PAIR_STRIDE + l15) * 2];

        #pragma unroll
        for (int t = 0; t < TILES_PER_WAVE; ++t) {
            v2f bfrag = *(const v2f*)(brow + t * 32);    // t*16 cols * 2 floats
            acc[t] = __builtin_amdgcn_wmma_f32_16x16x4_f32(
                         /*neg_a=*/false, afrag,
                         /*neg_b=*/false, bfrag,
                         /*c_mod=*/(short)0, acc[t],
                         /*reuse_a=*/false, /*reuse_b=*/false);
        }
    }

    // ---- Gabor activation epilogue ---------------------------------------
    // C/D layout: VGPR j, lane L -> M = j + 8*(L>>4), N = L&15.
    #pragma unroll
    for (int t = 0; t < TILES_PER_WAVE; ++t) {
        const int col = colBase + t * 16 + l15;
        #pragma unroll
        for (int j = 0; j < 8; ++j) {
            const float c   = acc[t][j];
            const int   row = rowBase + j + 8 * hg;
            float s, co;
            __sincosf(OMEGA0 * c, &s, &co);
            const float env = __expf(-SCALE0_SQ * c * c);
            float2 o2 = make_float2(env * co, env * s);
            // out[n, o, 2]: lanes 0-15 write 128 contiguous bytes per row.
            *(float2*)(out + ((size_t)row * OUT_DIM + col) * 2) = o2;
        }
    }
}

extern "C" void kernel_launch(void* const* d_in, const int* in_sizes, int n_in,
                              void* d_out, int out_size, void* d_ws, size_t ws_size,
                              hipStream_t stream) {
    const float* x    = (const float*)d_in[0];   // [N, 256]
    const float* W    = (const float*)d_in[1];   // [256, 256]
    const float* bias = (const float*)d_in[2];   // [256]
    float*       out  = (float*)d_out;           // [N, 256, 2]

    const int n_rows = in_sizes[0] / K_DIM;      // 262144
    const int blocks = (n_rows / ROWS_PER_WG) * 2;   // 4096 (row-block x col-half)

    // > 64 KB dynamic LDS (147456 B): raise the limit (host-side, capture-safe).
    (void)hipFuncSetAttribute((const void*)gabor_wmma_kernel,
                              hipFuncAttributeMaxDynamicSharedMemorySize,
                              (int)LDS_BYTES);

    gabor_wmma_kernel<<<blocks, WAVES_PER_WG * 32, LDS_BYTES, stream>>>(x, W, bias, out);
}